// DCN_60584808677886
// MI455X (gfx1250) — compile-verified
//
#include <hip/hip_runtime.h>
#include <math.h>

// ---------------------------------------------------------------------------
// Fused DCN inference for MI455X (gfx1250, wave32, WMMA).
// All per-row work that touches x is collapsed into ONE [B,1024]x[1024,80]
// fp32 GEMM (cols: 64 of w1 | cw1 cw2 cw3 | w_out head | 12 pad), executed
// with V_WMMA_F32_16X16X4_F32. Deep-tower tail (64->48->24) also uses WMMA.
// x is streamed through LDS with async global->LDS copies, double buffered.
// ---------------------------------------------------------------------------

typedef float v2f __attribute__((ext_vector_type(2)));
typedef float v4f __attribute__((ext_vector_type(4)));
typedef float v8f __attribute__((ext_vector_type(8)));
typedef int   v4i __attribute__((ext_vector_type(4)));

typedef __attribute__((address_space(1))) v4i gv4i;  // global int4
typedef __attribute__((address_space(3))) v4i lv4i;  // LDS int4

#define BN_EPS 1e-3f

// ---- workspace layout (float offsets) ----
#define WPP_OFF   0          // packed big weights: 512 pairs * 160 = 81920
#define W2P_OFF   81920      // packed w2: 32 pairs * 96 = 3072
#define W3P_OFF   84992      // packed w3 (padded to 32 cols): 24 pairs * 64 = 1536
#define CONST_OFF 86528      // 436 floats of folded constants
// const sub-offsets inside CONST block
#define CB1 0
#define CS1 64
#define CT1 128
#define CB2 192
#define CS2 240
#define CT2 288
#define CB3 336
#define CS3 360
#define CT3 384
#define CWT 408
#define CSC 432
#define CONST_N 436

// ---- CDNA5 async global->LDS path (guarded; falls back to sync copy) ----
#if defined(__has_builtin)
#if __has_builtin(__builtin_amdgcn_global_load_async_to_lds_b128) && \
    __has_builtin(__builtin_amdgcn_s_wait_asynccnt)
#define DCN_ASYNC 1
#endif
#endif
#ifndef DCN_ASYNC
#define DCN_ASYNC 0
#endif

__device__ __forceinline__ void cp16_g2l(const float* g, float* l) {
#if DCN_ASYNC
  __builtin_amdgcn_global_load_async_to_lds_b128((gv4i*)g, (lv4i*)l, 0, 0);
#else
  *(v4f*)l = *(const v4f*)g;
#endif
}

__device__ __forceinline__ void wait_async_done() {
#if DCN_ASYNC
  __builtin_amdgcn_s_wait_asynccnt(0);
#endif
}

// D = A(16x4) * B(4x16) + C, fp32. 8-arg form (neg_a,A,neg_b,B,cmod,C,ra,rb)
__device__ __forceinline__ v8f wmma4(v2f a, v2f b, v8f c) {
  return __builtin_amdgcn_wmma_f32_16x16x4_f32(false, a, false, b,
                                               (short)0, c, false, false);
}

// ---------------------------------------------------------------------------
// Prep kernel: pack weights into WMMA B-fragment "pair" layout and fold BN.
// B-matrix 4x16 layout: lane l holds col n=l&15, k-half kh=l>>4; VGPR0/1 are
// consecutive k -> store W[k][n],W[k+1][n] adjacent: P[(k>>1)*2N + n*2 + (k&1)]
// ---------------------------------------------------------------------------
__global__ __launch_bounds__(256) void dcn_prep(
    const float* __restrict__ cross_w, const float* __restrict__ cross_b,
    const float* __restrict__ w1, const float* __restrict__ b1,
    const float* __restrict__ g1, const float* __restrict__ be1,
    const float* __restrict__ m1, const float* __restrict__ v1,
    const float* __restrict__ w2, const float* __restrict__ b2,
    const float* __restrict__ g2, const float* __restrict__ be2,
    const float* __restrict__ m2, const float* __restrict__ v2,
    const float* __restrict__ w3, const float* __restrict__ b3,
    const float* __restrict__ g3, const float* __restrict__ be3,
    const float* __restrict__ m3, const float* __restrict__ v3,
    const float* __restrict__ w_out, const float* __restrict__ b_out,
    float* __restrict__ ws) {
  const int tid = threadIdx.x;

  // big packed matrix: 1024 x 80  (cols 0..63 = w1, 64..66 = cw1..3, 67 = wout head)
  for (int idx = tid; idx < 1024 * 80; idx += 256) {
    int k = idx / 80, n = idx % 80;
    float val;
    if (n < 64)       val = w1[k * 64 + n];
    else if (n < 67)  val = cross_w[(n - 64) * 1024 + k];
    else if (n == 67) val = w_out[k];
    else              val = 0.0f;
    ws[WPP_OFF + (k >> 1) * 160 + n * 2 + (k & 1)] = val;
  }
  // w2: 64 x 48
  for (int idx = tid; idx < 64 * 48; idx += 256) {
    int k = idx / 48, n = idx % 48;
    ws[W2P_OFF + (k >> 1) * 96 + n * 2 + (k & 1)] = w2[k * 48 + n];
  }
  // w3: 48 x 24, zero-padded to 32 cols
  for (int idx = tid; idx < 48 * 32; idx += 256) {
    int k = idx / 32, n = idx % 32;
    ws[W3P_OFF + (k >> 1) * 64 + n * 2 + (k & 1)] = (n < 24) ? w3[k * 24 + n] : 0.0f;
  }
  // BN folds: BN(z) = z*scale + shift,  scale = g*rsqrt(v+eps), shift = be - m*scale
  for (int j = tid; j < 64; j += 256) {
    float sc = g1[j] * rsqrtf(v1[j] + BN_EPS);
    ws[CONST_OFF + CB1 + j] = b1[j];
    ws[CONST_OFF + CS1 + j] = sc;
    ws[CONST_OFF + CT1 + j] = be1[j] - m1[j] * sc;
  }
  for (int j = tid; j < 48; j += 256) {
    float sc = g2[j] * rsqrtf(v2[j] + BN_EPS);
    ws[CONST_OFF + CB2 + j] = b2[j];
    ws[CONST_OFF + CS2 + j] = sc;
    ws[CONST_OFF + CT2 + j] = be2[j] - m2[j] * sc;
  }
  for (int j = tid; j < 24; j += 256) {
    float sc = g3[j] * rsqrtf(v3[j] + BN_EPS);
    ws[CONST_OFF + CB3 + j] = b3[j];
    ws[CONST_OFF + CS3 + j] = sc;
    ws[CONST_OFF + CT3 + j] = be3[j] - m3[j] * sc;
    ws[CONST_OFF + CWT + j] = w_out[1024 + j];
  }
  // scalar constants: c12 = cb1.cw2, c123 = (cb1+cb2).cw3, c_out = (cb1+cb2+cb3).wh
  __shared__ float red[3][256];
  float p12 = 0.f, p123 = 0.f, pout = 0.f;
  for (int d = tid; d < 1024; d += 256) {
    float a = cross_b[d], b = cross_b[1024 + d], c = cross_b[2048 + d];
    p12  += a * cross_w[1024 + d];
    p123 += (a + b) * cross_w[2048 + d];
    pout += (a + b + c) * w_out[d];
  }
  red[0][tid] = p12; red[1][tid] = p123; red[2][tid] = pout;
  __syncthreads();
  for (int s = 128; s > 0; s >>= 1) {
    if (tid < s) {
      red[0][tid] += red[0][tid + s];
      red[1][tid] += red[1][tid + s];
      red[2][tid] += red[2][tid + s];
    }
    __syncthreads();
  }
  if (tid == 0) {
    ws[CONST_OFF + CSC + 0] = red[0][0];
    ws[CONST_OFF + CSC + 1] = red[1][0];
    ws[CONST_OFF + CSC + 2] = red[2][0];
    ws[CONST_OFF + CSC + 3] = b_out[0];
  }
}

// ---------------------------------------------------------------------------
// Main fused kernel: 128 blocks x 256 threads (8 wave32s). Wave = 16 rows.
// K chunked by 64, x-tile + W-chunk double-buffered in LDS via async copies.
// ---------------------------------------------------------------------------
__device__ __forceinline__ void issue_chunk(const float* __restrict__ x,
                                            const float* __restrict__ wsf,
                                            int rowbase, int kc,
                                            float* xsb, float* wpb, int tid) {
  // x tile: 128 rows x 64 cols = 2048 float4
#pragma unroll
  for (int i = 0; i < 8; ++i) {
    int f = tid + 256 * i;
    int r = f >> 4, s = f & 15;
    cp16_g2l(x + (size_t)(rowbase + r) * 1024 + kc * 64 + s * 4,
             xsb + r * 64 + s * 4);
  }
  // W chunk: 32 pairs * 160 = 5120 floats = 1280 float4 (contiguous)
#pragma unroll
  for (int i = 0; i < 5; ++i) {
    int f = tid + 256 * i;
    cp16_g2l(wsf + WPP_OFF + (size_t)kc * 5120 + f * 4, wpb + f * 4);
  }
}

__global__ __launch_bounds__(256) void dcn_main(const float* __restrict__ x,
                                                const float* __restrict__ wsf,
                                                float* __restrict__ out) {
  __shared__ __align__(16) float xs[2][128 * 64];   // 64 KB
  __shared__ __align__(16) float wps[2][5120];      // 40 KB
  __shared__ __align__(16) float w2s[3072];         // 12 KB
  __shared__ __align__(16) float w3s[1536];         //  6 KB
  __shared__ __align__(16) float cs[CONST_N];       // ~1.7 KB

  const int tid  = threadIdx.x;
  const int lane = tid & 31;
  const int wv   = tid >> 5;
  const int rowbase = blockIdx.x * 128;

  // stage small constants / tail weights
  for (int i = tid; i < 3072; i += 256) w2s[i] = wsf[W2P_OFF + i];
  for (int i = tid; i < 1536; i += 256) w3s[i] = wsf[W3P_OFF + i];
  for (int i = tid; i < CONST_N; i += 256) cs[i] = wsf[CONST_OFF + i];

  issue_chunk(x, wsf, rowbase, 0, xs[0], wps[0], tid);

  v8f acc[5];
#pragma unroll
  for (int t = 0; t < 5; ++t) acc[t] = (v8f){0, 0, 0, 0, 0, 0, 0, 0};

  const int m  = lane & 15;   // A row / C col within tile
  const int kh = lane >> 4;   // k-half select
  const int arow = (wv * 16 + m) * 64 + 2 * kh;
  const int bcol = m * 2;

  int buf = 0;
  for (int kc = 0; kc < 16; ++kc) {
    wait_async_done();
    __syncthreads();
    if (kc + 1 < 16)
      issue_chunk(x, wsf, rowbase, kc + 1, xs[buf ^ 1], wps[buf ^ 1], tid);
    const float* xb = xs[buf];
    const float* wb = wps[buf];
#pragma unroll
    for (int ks = 0; ks < 64; ks += 4) {
      v2f a = *(const v2f*)(xb + arow + ks);
      const float* bp = wb + ((ks >> 1) + kh) * 160 + bcol;
#pragma unroll
      for (int t = 0; t < 5; ++t) {
        v2f b = *(const v2f*)(bp + t * 32);
        acc[t] = wmma4(a, b, acc[t]);
      }
    }
    buf ^= 1;
  }
  __syncthreads();  // main-loop LDS retired; reuse as epilogue scratch

  // per-wave scratch: 2560 floats = [y 16x80 | h2 16x48 | h3 16x24]
  float* xsflat = &xs[0][0];   // 16384 floats
  float* wpflat = &wps[0][0];  // 10240 floats
  float* ys = (wv < 6) ? (xsflat + wv * 2560) : (wpflat + (wv - 6) * 2560);
  float* h2 = ys + 1280;
  float* h3 = ys + 2048;

  // 1) dump accumulators: D layout -> lane l holds col n=l&15, rows v+8*kh
  const int m0 = 8 * kh;
#pragma unroll
  for (int t = 0; t < 5; ++t)
#pragma unroll
    for (int v = 0; v < 8; ++v)
      ys[(v + m0) * 80 + t * 16 + m] = acc[t][v];
  __syncthreads();

  // 2) h1 = BN1(relu(y + b1)) in place on cols 0..63 (cols 64..67 keep d1..d4)
#pragma unroll
  for (int i = 0; i < 32; ++i) {
    int idx = lane + 32 * i;           // 0..1023
    int col = idx & 63, row = idx >> 6;
    float y = ys[row * 80 + col] + cs[CB1 + col];
    y = fmaxf(y, 0.0f);
    ys[row * 80 + col] = y * cs[CS1 + col] + cs[CT1 + col];
  }
  __syncthreads();

  // 3) h2 = BN2(tanh(h1 @ w2 + b2)) : K=64, N=48 -> 16 ksteps x 3 tiles
  v8f a2[3];
#pragma unroll
  for (int t = 0; t < 3; ++t) a2[t] = (v8f){0, 0, 0, 0, 0, 0, 0, 0};
#pragma unroll
  for (int ks = 0; ks < 64; ks += 4) {
    v2f a = *(const v2f*)&ys[m * 80 + ks + 2 * kh];
#pragma unroll
    for (int t = 0; t < 3; ++t) {
      v2f b = *(const v2f*)&w2s[((ks >> 1) + kh) * 96 + (t * 16 + m) * 2];
      a2[t] = wmma4(a, b, a2[t]);
    }
  }
#pragma unroll
  for (int t = 0; t < 3; ++t) {
    int col = t * 16 + m;
#pragma unroll
    for (int v = 0; v < 8; ++v) {
      float z = tanhf(a2[t][v] + cs[CB2 + col]);
      h2[(v + m0) * 48 + col] = z * cs[CS2 + col] + cs[CT2 + col];
    }
  }
  __syncthreads();

  // 4) h3 = BN3(tanh(h2 @ w3 + b3)) : K=48, N=24(pad 32) -> 12 ksteps x 2 tiles
  v8f a3[2];
#pragma unroll
  for (int t = 0; t < 2; ++t) a3[t] = (v8f){0, 0, 0, 0, 0, 0, 0, 0};
#pragma unroll
  for (int ks = 0; ks < 48; ks += 4) {
    v2f a = *(const v2f*)&h2[m * 48 + ks + 2 * kh];
#pragma unroll
    for (int t = 0; t < 2; ++t) {
      v2f b = *(const v2f*)&w3s[((ks >> 1) + kh) * 64 + (t * 16 + m) * 2];
      a3[t] = wmma4(a, b, a3[t]);
    }
  }
#pragma unroll
  for (int t = 0; t < 2; ++t) {
    int col = t * 16 + m;
    if (col < 24) {
#pragma unroll
      for (int v = 0; v < 8; ++v) {
        float z = tanhf(a3[t][v] + cs[CB3 + col]);
        h3[(v + m0) * 24 + col] = z * cs[CS3 + col] + cs[CT3 + col];
      }
    }
  }
  __syncthreads();

  // 5) cross-net scalars + output head: one lane per row
  if (lane < 16) {
    int row = lane;
    const float d1 = ys[row * 80 + 64], d2 = ys[row * 80 + 65];
    const float d3 = ys[row * 80 + 66], d4 = ys[row * 80 + 67];
    const float c12 = cs[CSC + 0], c123 = cs[CSC + 1];
    const float cout = cs[CSC + 2], bout = cs[CSC + 3];
    float s1 = d1, t1 = 1.0f + s1;
    float s2 = t1 * d2 + c12, t2 = t1 + s2;
    float s3 = t2 * d3 + c123, alpha = t2 + s3;
    float dot = 0.0f;
#pragma unroll
    for (int j = 0; j < 24; ++j) dot += h3[row * 24 + j] * cs[CWT + j];
    float z = alpha * d4 + cout + dot + bout;
    out[rowbase + wv * 16 + row] = 1.0f / (1.0f + __expf(-z));
  }
}

// ---------------------------------------------------------------------------
extern "C" void kernel_launch(void* const* d_in, const int* in_sizes, int n_in,
                              void* d_out, int out_size, void* d_ws, size_t ws_size,
                              hipStream_t stream) {
  (void)in_sizes; (void)n_in; (void)out_size; (void)ws_size;
  const float* x       = (const float*)d_in[0];
  const float* cross_w = (const float*)d_in[1];
  const float* cross_b = (const float*)d_in[2];
  const float* w1   = (const float*)d_in[3];
  const float* b1   = (const float*)d_in[4];
  const float* g1   = (const float*)d_in[5];
  const float* be1  = (const float*)d_in[6];
  const float* m1   = (const float*)d_in[7];
  const float* v1   = (const float*)d_in[8];
  const float* w2   = (const float*)d_in[9];
  const float* b2   = (const float*)d_in[10];
  const float* g2   = (const float*)d_in[11];
  const float* be2  = (const float*)d_in[12];
  const float* m2   = (const float*)d_in[13];
  const float* v2   = (const float*)d_in[14];
  const float* w3   = (const float*)d_in[15];
  const float* b3   = (const float*)d_in[16];
  const float* g3   = (const float*)d_in[17];
  const float* be3  = (const float*)d_in[18];
  const float* m3   = (const float*)d_in[19];
  const float* v3   = (const float*)d_in[20];
  const float* wout = (const float*)d_in[21];
  const float* bout = (const float*)d_in[22];
  float* wsf = (float*)d_ws;

  dcn_prep<<<1, 256, 0, stream>>>(cross_w, cross_b, w1, b1, g1, be1, m1, v1,
                                  w2, b2, g2, be2, m2, v2,
                                  w3, b3, g3, be3, m3, v3, wout, bout, wsf);
  dcn_main<<<128, 256, 0, stream>>>(x, wsf, (float*)d_out);
}